// S4Enrichment1_55095840473855
// MI455X (gfx1250) — compile-verified
//
#include <hip/hip_runtime.h>
#include <cmath>

// Problem constants (from reference)
#define Hc   1024   // channels
#define Nc   64     // SSM state size
#define Lc   4096   // sequence length = 16*16*16
#define OUTc 512    // output features
#define BSc  2      // batch
#define KPAD 4608   // padded K row: 255 zeros | 4096 kernel | 257 tail zeros

typedef float v2f __attribute__((ext_vector_type(2)));
typedef float v8f __attribute__((ext_vector_type(8)));

// ---------------------------------------------------------------------------
// Kernel 1: S4D ZOH discretization.  Per (h,n): dtA = dt*A ; Ct = C*(exp(dtA)-1)/A
// ---------------------------------------------------------------------------
__global__ void __launch_bounds__(256)
s4_discretize(const float* __restrict__ log_dt,
              const float* __restrict__ A_re, const float* __restrict__ A_im,
              const float* __restrict__ C_re, const float* __restrict__ C_im,
              float* __restrict__ ctre, float* __restrict__ ctim,
              float* __restrict__ dare, float* __restrict__ daim) {
    int idx = blockIdx.x * blockDim.x + threadIdx.x;
    if (idx >= Hc * Nc) return;
    int h = idx >> 6;
    float dt = __expf(log_dt[h]);
    float ar = A_re[idx], ai = A_im[idx];
    float cr = C_re[idx], ci = C_im[idx];
    float dr = dt * ar, di = dt * ai;
    float s, c;
    float sc = __expf(dr);
    __sincosf(di, &s, &c);
    float m_re = sc * c - 1.0f, m_im = sc * s;
    float inv = 1.0f / (ar * ar + ai * ai);           // A_re = -0.5 -> never 0
    float q_re = (m_re * ar + m_im * ai) * inv;       // (exp(dtA)-1)/A
    float q_im = (m_im * ar - m_re * ai) * inv;
    ctre[idx] = cr * q_re - ci * q_im;
    ctim[idx] = cr * q_im + ci * q_re;
    dare[idx] = dr;
    daim[idx] = di;
}

// ---------------------------------------------------------------------------
// Kernel 2: Kpad[h, 255+l] = 2*Re( sum_n Ct[h,n] * exp(dtA[h,n]*l) )
// Row is zero-padded (255 left, 257 right) so the conv kernel's async tile
// loads need no causality predicate.
// ---------------------------------------------------------------------------
__global__ void __launch_bounds__(256)
s4_build_K(const float* __restrict__ ctre, const float* __restrict__ ctim,
           const float* __restrict__ dare, const float* __restrict__ daim,
           float* __restrict__ Kb) {
    __shared__ float sctr[Nc], scti[Nc], sdar[Nc], sdai[Nc];
    int h = blockIdx.y;
    int tid = threadIdx.x;
    if (tid < Nc) {
        int g = h * Nc + tid;
        sctr[tid] = ctre[g]; scti[tid] = ctim[g];
        sdar[tid] = dare[g]; sdai[tid] = daim[g];
    }
    __syncthreads();
    int l = blockIdx.x * 256 + tid;
    float lf = (float)l;
    float acc = 0.0f;
#pragma unroll 4
    for (int n = 0; n < Nc; ++n) {
        float s, c;
        float scale = __expf(sdar[n] * lf);           // decays to 0, safe
        sincosf(sdai[n] * lf, &s, &c);                // precise: args up to ~8e4 rad
        acc = fmaf(scale, fmaf(sctr[n], c, -scti[n] * s), acc);
    }
    float* row = Kb + (size_t)h * KPAD;
    row[255 + l] = 2.0f * acc;
    if (blockIdx.x == 0 && tid < 255)  row[tid] = 0.0f;              // left pad
    if (blockIdx.x == 15 && tid < 257) row[255 + Lc + tid] = 0.0f;   // tail pad
}

// ---------------------------------------------------------------------------
// Kernel 3: V[b,c,l] = x[b,c,l] + posenc3d(l, c)   (reference transposes cancel)
// ---------------------------------------------------------------------------
__global__ void __launch_bounds__(256)
s4_posenc_add(const float* __restrict__ x, float* __restrict__ V) {
    size_t idx = (size_t)blockIdx.x * 256 + threadIdx.x;   // < BS*H*L
    int l = (int)(idx & (Lc - 1));
    int c = (int)((idx >> 12) & (Hc - 1));
    int pos, ac;
    if (c < 342)      { pos = l >> 8;        ac = c; }        // f axis
    else if (c < 684) { pos = (l >> 4) & 15; ac = c - 342; }  // h axis
    else              { pos = l & 15;        ac = c - 684; }  // w axis
    int i2 = ac & ~1;
    float freq = __expf((float)i2 * (-9.210340371976184f / 342.0f)); // 10000^(-i2/342)
    float s = (float)pos * freq;                                      // |s| <= 15
    float pe = (ac & 1) ? cosf(s) : sinf(s);
    V[idx] = x[idx] + pe;
}

// ---------------------------------------------------------------------------
// Kernel 4: causal depthwise conv + skip + tanh-GELU with CDNA5 async
// double-buffered LDS staging (GLOBAL_LOAD_ASYNC_TO_LDS_B32 / ASYNCcnt).
// Per stage each thread issues 3 async B32 loads (1 V, 2 padded-K); order is
// wait(asynccnt==0) -> workgroup barrier -> issue next stage -> compute, so
// next-stage loads overlap the 256-FMA inner loop without racing readers.
// ---------------------------------------------------------------------------
__global__ void __launch_bounds__(256)
s4_conv_gelu(const float* __restrict__ V, const float* __restrict__ Kb,
             const float* __restrict__ Dv, float* __restrict__ Y) {
    __shared__ float vsm[2][256];
    __shared__ float ksm[2][512];
    int tid = threadIdx.x;
    int ltile = blockIdx.x;            // 0..15
    int h = blockIdx.y;
    int b = blockIdx.z;
    const float* vrow = V + ((size_t)b * Hc + h) * Lc;
    const float* krow = Kb + (size_t)h * KPAD;   // padded: index (ltile-t)*256 + i

    auto issue_stage = [&](int t, int p) {
        unsigned lv = (unsigned)(uintptr_t)&vsm[p][tid];
        unsigned long long gv = (unsigned long long)(uintptr_t)(vrow + t * 256 + tid);
        asm volatile("global_load_async_to_lds_b32 %0, %1, off"
                     :: "v"(lv), "v"(gv) : "memory");
        int kb = (ltile - t) * 256;              // padded base, >= 0, max 4351 < KPAD
        unsigned lk0 = (unsigned)(uintptr_t)&ksm[p][tid];
        unsigned long long gk0 = (unsigned long long)(uintptr_t)(krow + kb + tid);
        asm volatile("global_load_async_to_lds_b32 %0, %1, off"
                     :: "v"(lk0), "v"(gk0) : "memory");
        unsigned lk1 = (unsigned)(uintptr_t)&ksm[p][tid + 256];
        unsigned long long gk1 = (unsigned long long)(uintptr_t)(krow + kb + 256 + tid);
        asm volatile("global_load_async_to_lds_b32 %0, %1, off"
                     :: "v"(lk1), "v"(gk1) : "memory");
    };

    issue_stage(0, 0);
    float acc = 0.0f;
    for (int t = 0; t <= ltile; ++t) {
        int p = t & 1;
        asm volatile("s_wait_asynccnt 0" ::: "memory");   // stage-t tiles landed (this wave)
        __syncthreads();                                  // all waves' tiles landed
        if (t < ltile) issue_stage(t + 1, p ^ 1);         // overlap with compute
#pragma unroll 8
        for (int j = 0; j < 256; ++j)
            acc = fmaf(vsm[p][j], ksm[p][tid - j + 255], acc);
    }
    // vsm[ltile&1] still holds the current tile of V
    float xv = acc + Dv[h] * vsm[ltile & 1][tid];
    float t3 = 0.7978845608028654f * fmaf(0.044715f * xv * xv, xv, xv);
    float g = 0.5f * xv * (1.0f + tanhf(t3));
    Y[((size_t)b * Hc + h) * Lc + ltile * 256 + tid] = g;
}

// ---------------------------------------------------------------------------
// Kernel 5: out[b,o,l] = sum_k Y[b,k,l] * W[k,o] + bias[o]
// GEMM M=8192, K=1024, N=512 with V_WMMA_F32_16X16X4_F32.
// One wave owns a 16x32 tile (two accumulators sharing each A fragment);
// A is read transposed from channel-major Y (coalesced along l); the
// epilogue transposes each 16x16 through LDS for contiguous-along-l stores.
// ---------------------------------------------------------------------------
__global__ void __launch_bounds__(256)
s4_out_gemm(const float* __restrict__ Y, const float* __restrict__ W,
            const float* __restrict__ bias, float* __restrict__ out) {
    __shared__ float tl[8][16 * 17];
    int wv   = threadIdx.x >> 5;
    int lane = threadIdx.x & 31;
    int tile = blockIdx.x * 8 + wv;          // 8192 tiles total
    int nt2 = tile & 15;                     // N/32 = 16
    int mt  = tile >> 4;                     // M/16 = 512
    int half = lane >> 4;
    int l16  = lane & 15;
    int mbase = mt * 16;
    int bidx  = mbase >> 12;                 // batch (L = 4096)
    int lbase = mbase & (Lc - 1);
    size_t aoff = (size_t)bidx * Hc * Lc + (size_t)(lbase + l16);
    int n0 = nt2 * 32 + l16;
    int n1 = n0 + 16;
    int kk = half * 2;                       // A 16x4 layout: lanes16-31 hold K+2,K+3
    v8f acc0 = {}, acc1 = {};
    for (int k0 = 0; k0 < Hc; k0 += 4) {
        if (k0 + 16 < Hc)                    // speculative prefetch of upcoming W rows
            __builtin_prefetch(W + (size_t)(k0 + 16) * OUTc + n0, 0, 0);
        v2f a, b0, b1;
        a.x  = Y[aoff + (size_t)(k0 + kk) * Lc];
        a.y  = Y[aoff + (size_t)(k0 + kk + 1) * Lc];
        b0.x = W[(size_t)(k0 + kk) * OUTc + n0];
        b0.y = W[(size_t)(k0 + kk + 1) * OUTc + n0];
        b1.x = W[(size_t)(k0 + kk) * OUTc + n1];
        b1.y = W[(size_t)(k0 + kk + 1) * OUTc + n1];
        acc0 = __builtin_amdgcn_wmma_f32_16x16x4_f32(false, a, false, b0,
                                                     (short)0, acc0, false, false);
        acc1 = __builtin_amdgcn_wmma_f32_16x16x4_f32(false, a, false, b1,
                                                     (short)0, acc1, false, false);
    }
    float* T = tl[wv];
#pragma unroll
    for (int tp = 0; tp < 2; ++tp) {
        v8f acc = tp ? acc1 : acc0;
        int ng = nt2 * 32 + tp * 16 + l16;
        float bn = bias[ng];
        __syncthreads();                     // uniform: previous readers done
#pragma unroll
        for (int r = 0; r < 8; ++r)          // C/D layout: VGPR r -> M = r + half*8
            T[(r + half * 8) * 17 + l16] = acc[r] + bn;
        __syncthreads();
        size_t obase = (size_t)bidx * OUTc * Lc + (size_t)ng * Lc
                     + (size_t)(lbase + half * 8);
#pragma unroll
        for (int i = 0; i < 8; ++i)          // contiguous along l
            out[obase + i] = T[(half * 8 + i) * 17 + l16];
    }
}

// ---------------------------------------------------------------------------
extern "C" void kernel_launch(void* const* d_in, const int* in_sizes, int n_in,
                              void* d_out, int out_size, void* d_ws, size_t ws_size,
                              hipStream_t stream) {
    const float* x      = (const float*)d_in[0];
    const float* log_dt = (const float*)d_in[1];
    const float* A_re   = (const float*)d_in[2];
    const float* A_im   = (const float*)d_in[3];
    const float* C_re   = (const float*)d_in[4];
    const float* C_im   = (const float*)d_in[5];
    const float* Dv     = (const float*)d_in[6];
    const float* W_out  = (const float*)d_in[7];
    const float* b_out  = (const float*)d_in[8];
    float* out = (float*)d_out;

    // workspace (floats): Ct/dtA (4x65536) | Kpad (H*4608) | V (BS*H*L) | Y (BS*H*L)
    float* ws   = (float*)d_ws;
    float* ctre = ws;
    float* ctim = ctre + Hc * Nc;
    float* dare = ctim + Hc * Nc;
    float* daim = dare + Hc * Nc;
    float* Kb   = daim + Hc * Nc;                       // 18 MB (padded)
    float* Vb   = Kb + (size_t)Hc * KPAD;               // 33.5 MB
    float* Yb   = Vb + (size_t)BSc * Hc * Lc;           // 33.5 MB

    s4_discretize<<<dim3(Hc * Nc / 256), dim3(256), 0, stream>>>(
        log_dt, A_re, A_im, C_re, C_im, ctre, ctim, dare, daim);

    s4_build_K<<<dim3(Lc / 256, Hc), dim3(256), 0, stream>>>(ctre, ctim, dare, daim, Kb);

    s4_posenc_add<<<dim3((unsigned)((size_t)BSc * Hc * Lc / 256)), dim3(256), 0, stream>>>(x, Vb);

    s4_conv_gelu<<<dim3(Lc / 256, Hc, BSc), dim3(256), 0, stream>>>(Vb, Kb, Dv, Yb);

    s4_out_gemm<<<dim3((BSc * Lc / 16) * (OUTc / 32) / 8), dim3(256), 0, stream>>>(
        Yb, W_out, b_out, out);
}